// LRGCCF_73237782331839
// MI455X (gfx1250) — compile-verified
//
#include <hip/hip_runtime.h>
#include <hip/hip_bf16.h>
#include <math.h>

#define USER_NUM  100000
#define ITEM_NUM  50000
#define FACTOR    64
#define N_EDGES   3200000
#define BATCH     4096
#define LAMADA    1.0e-4f

typedef __attribute__((ext_vector_type(2))) float v2f;
typedef __attribute__((ext_vector_type(8))) float v8f;

// ---------------------------------------------------------------------------
// 0) zero the scalar accumulators (must happen every call: graph replays do
//    not re-poison d_ws, and we accumulate atomically into these).
__global__ void lrgccf_zero_acc(float* acc) {
    if (threadIdx.x < 8) acc[threadIdx.x] = 0.0f;
}

// ---------------------------------------------------------------------------
// 1) dst[row, :] = src[row, :] * d[row]    (row-major, FACTOR=64)
//    One thread per float4 => 16 threads per row, fully coalesced B128 ops.
__global__ void lrgccf_scale_rows(const float* __restrict__ src,
                                  const float* __restrict__ d,
                                  float* __restrict__ dst, int rows) {
    int t = blockIdx.x * blockDim.x + threadIdx.x;      // float4 index
    int total = rows * (FACTOR / 4);
    if (t >= total) return;
    int row = t >> 4;
    float4 v = ((const float4*)src)[t];
    float s = d[row];
    v.x *= s; v.y *= s; v.z *= s; v.w *= s;
    ((float4*)dst)[t] = v;
}

// ---------------------------------------------------------------------------
// 2) Fused bipartite scatter:
//      dst_u[u_idx[e]] += src_i[i_idx[e]] * ui_vals[e]
//      dst_i[i_idx[e]] += src_u[u_idx[e]] * iu_vals[e]
//    16 threads per edge, each handles one float4 chunk (256B contiguous per
//    row -> full cachelines).  Native f32 no-return atomics resolve in L2;
//    the whole 115MB working set fits in the 192MB L2.
__global__ void lrgccf_spmm(const float* __restrict__ src_u,
                            const float* __restrict__ src_i,
                            float* __restrict__ dst_u,
                            float* __restrict__ dst_i,
                            const int* __restrict__ u_idx,
                            const int* __restrict__ i_idx,
                            const float* __restrict__ ui_vals,
                            const float* __restrict__ iu_vals,
                            int n_edges) {
    int t = blockIdx.x * blockDim.x + threadIdx.x;
    int e = t >> 4;
    if (e >= n_edges) return;
    int c = t & 15;                       // float4 chunk within the row
    int ur = u_idx[e];
    int ir = i_idx[e];
    float uv = ui_vals[e];
    float iv = iu_vals[e];

    const float4* irow = (const float4*)(src_i + (size_t)ir * FACTOR);
    const float4* urow = (const float4*)(src_u + (size_t)ur * FACTOR);
    float4 a = irow[c];
    float4 b = urow[c];

    float* du = dst_u + (size_t)ur * FACTOR + c * 4;
    float* di = dst_i + (size_t)ir * FACTOR + c * 4;
    unsafeAtomicAdd(du + 0, a.x * uv);
    unsafeAtomicAdd(du + 1, a.y * uv);
    unsafeAtomicAdd(du + 2, a.z * uv);
    unsafeAtomicAdd(du + 3, a.w * uv);
    unsafeAtomicAdd(di + 0, b.x * iv);
    unsafeAtomicAdd(di + 1, b.y * iv);
    unsafeAtomicAdd(di + 2, b.z * iv);
    unsafeAtomicAdd(di + 3, b.w * iv);
}

// ---------------------------------------------------------------------------
// 3) Batch loss with V_WMMA_F32_16X16X4_F32.
//    One wave (32 lanes) per 16 batch rows.  A = u-tile (16xK), B = p^T
//    (Kx16); D[m][n] = sum_k u[m,k]*p[n,k]; diagonal => per-row dot product.
//    ISA layouts (16x4 f32 A; 4x16 f32 B mirrors it): lane L supplies
//    row/col L%16, k-pair {0,1} for lanes 0-15 and {2,3} for lanes 16-31.
//    u-rows are composed on the fly as e + g1 + g2 (gcn_u never materialized).
__global__ __launch_bounds__(32)
void lrgccf_batch_loss(const float* __restrict__ eu,
                       const float* __restrict__ g1u,
                       const float* __restrict__ g2u,
                       const float* __restrict__ ei,
                       const float* __restrict__ g1i,
                       const float* __restrict__ g2i,
                       const int* __restrict__ user,
                       const int* __restrict__ item_i,
                       const int* __restrict__ item_j,
                       float* __restrict__ acc) {
    const int lane = threadIdx.x;
    const int r    = lane & 15;                 // row within tile
    const int kh   = (lane < 16) ? 0 : 2;      // k half within group of 4
    const int b    = blockIdx.x * 16 + r;      // batch row this lane feeds

    const size_t ur = (size_t)user[b]   * FACTOR;
    const size_t ii = (size_t)item_i[b] * FACTOR;
    const size_t ij = (size_t)item_j[b] * FACTOR;

    v8f ci = {0.f,0.f,0.f,0.f,0.f,0.f,0.f,0.f};
    v8f cj = {0.f,0.f,0.f,0.f,0.f,0.f,0.f,0.f};
    float su = 0.f, spi = 0.f, spj = 0.f;      // square-sum partials

#pragma unroll
    for (int kk = 0; kk < 16; ++kk) {
        const int k = kk * 4 + kh;
        // u = eu + g1u + g2u   (two consecutive k values -> float2 loads)
        float2 ua = *(const float2*)(eu  + ur + k);
        float2 ub = *(const float2*)(g1u + ur + k);
        float2 uc = *(const float2*)(g2u + ur + k);
        v2f a; a.x = ua.x + ub.x + uc.x; a.y = ua.y + ub.y + uc.y;

        float2 pa = *(const float2*)(ei  + ii + k);
        float2 pb = *(const float2*)(g1i + ii + k);
        float2 pc = *(const float2*)(g2i + ii + k);
        v2f bi; bi.x = pa.x + pb.x + pc.x; bi.y = pa.y + pb.y + pc.y;

        float2 qa = *(const float2*)(ei  + ij + k);
        float2 qb = *(const float2*)(g1i + ij + k);
        float2 qc = *(const float2*)(g2i + ij + k);
        v2f bj; bj.x = qa.x + qb.x + qc.x; bj.y = qa.y + qb.y + qc.y;

        su  += a.x  * a.x  + a.y  * a.y;       // each (row,k) hit exactly once
        spi += bi.x * bi.x + bi.y * bi.y;
        spj += bj.x * bj.x + bj.y * bj.y;

        // D = A x B + C, exact f32 MACs (16x16x4)
        ci = __builtin_amdgcn_wmma_f32_16x16x4_f32(false, a, false, bi,
                                                   (short)0, ci, false, false);
        cj = __builtin_amdgcn_wmma_f32_16x16x4_f32(false, a, false, bj,
                                                   (short)0, cj, false, false);
    }

    // ---- diagonal extraction: (m,m) lives at VGPR m, lane m (m<8) and
    //      VGPR m-8, lane m+16 (m>=8) -> lanes 0..7 and 24..31.
    const bool have = (lane < 8) || (lane >= 24);
    const int  idx  = (lane < 8) ? lane : (lane - 24);   // which C component
    float di = 0.f, dj = 0.f;
#pragma unroll
    for (int q = 0; q < 8; ++q) {
        if (idx == q) { di = ci[q]; dj = cj[q]; }
    }

    float sp = 0.f;
    if (have) {
        float x = dj - di;                       // -(pred_i - pred_j)
        sp = fmaxf(x, 0.f) + log1pf(expf(-fabsf(x)));   // stable softplus
    }

    // ---- wave32 reductions
#pragma unroll
    for (int off = 16; off > 0; off >>= 1) {
        sp  += __shfl_xor(sp,  off, 32);
        su  += __shfl_xor(su,  off, 32);
        spi += __shfl_xor(spi, off, 32);
        spj += __shfl_xor(spj, off, 32);
    }
    if (lane == 0) {
        unsafeAtomicAdd(&acc[0], sp);
        unsafeAtomicAdd(&acc[1], su);
        unsafeAtomicAdd(&acc[2], spi);
        unsafeAtomicAdd(&acc[3], spj);
    }
}

// ---------------------------------------------------------------------------
// 4) finalize scalar loss
__global__ void lrgccf_finalize(const float* __restrict__ acc,
                                float* __restrict__ out) {
    const float inv_b  = 1.0f / (float)BATCH;
    const float inv_be = 1.0f / ((float)BATCH * (float)FACTOR);
    float loss2 = acc[0] * inv_b;
    float l2 = LAMADA * (acc[1] * inv_be) + LAMADA * ((acc[2] + acc[3]) * inv_be);
    out[0] = loss2 + l2;
}

// ---------------------------------------------------------------------------
extern "C" void kernel_launch(void* const* d_in, const int* in_sizes, int n_in,
                              void* d_out, int out_size, void* d_ws, size_t ws_size,
                              hipStream_t stream) {
    const float* embed_user = (const float*)d_in[0];
    const float* embed_item = (const float*)d_in[1];
    const int*   u_idx      = (const int*)d_in[2];
    const int*   i_idx      = (const int*)d_in[3];
    const float* ui_vals    = (const float*)d_in[4];
    const float* iu_vals    = (const float*)d_in[5];
    const float* d_i        = (const float*)d_in[6];
    const float* d_j        = (const float*)d_in[7];
    const int*   user       = (const int*)d_in[8];
    const int*   item_i     = (const int*)d_in[9];
    const int*   item_j     = (const int*)d_in[10];
    float* out = (float*)d_out;

    // workspace layout: [acc(256B pad)] [g1u] [g1i] [g2u] [g2i]
    char* ws   = (char*)d_ws;
    float* acc = (float*)ws;
    float* g1u = (float*)(ws + 256);
    float* g1i = g1u + (size_t)USER_NUM * FACTOR;
    float* g2u = g1i + (size_t)ITEM_NUM * FACTOR;
    float* g2i = g2u + (size_t)USER_NUM * FACTOR;

    const int TPB = 256;
    const int u_v4 = USER_NUM * (FACTOR / 4);
    const int i_v4 = ITEM_NUM * (FACTOR / 4);
    const int edge_threads = N_EDGES * 16;

    lrgccf_zero_acc<<<1, 32, 0, stream>>>(acc);

    // layer 1: g1u = eu*di + A^T(ei*ui), g1i = ei*dj + A(eu*iu)
    lrgccf_scale_rows<<<(u_v4 + TPB - 1) / TPB, TPB, 0, stream>>>(embed_user, d_i, g1u, USER_NUM);
    lrgccf_scale_rows<<<(i_v4 + TPB - 1) / TPB, TPB, 0, stream>>>(embed_item, d_j, g1i, ITEM_NUM);
    lrgccf_spmm<<<(edge_threads + TPB - 1) / TPB, TPB, 0, stream>>>(
        embed_user, embed_item, g1u, g1i, u_idx, i_idx, ui_vals, iu_vals, N_EDGES);

    // layer 2: g2u = g1u*di + scatter(g1i), g2i = g1i*dj + scatter(g1u)
    lrgccf_scale_rows<<<(u_v4 + TPB - 1) / TPB, TPB, 0, stream>>>(g1u, d_i, g2u, USER_NUM);
    lrgccf_scale_rows<<<(i_v4 + TPB - 1) / TPB, TPB, 0, stream>>>(g1i, d_j, g2i, ITEM_NUM);
    lrgccf_spmm<<<(edge_threads + TPB - 1) / TPB, TPB, 0, stream>>>(
        g1u, g1i, g2u, g2i, u_idx, i_idx, ui_vals, iu_vals, N_EDGES);

    // loss: one wave per 16 batch rows (BATCH/16 = 256 waves)
    lrgccf_batch_loss<<<BATCH / 16, 32, 0, stream>>>(
        embed_user, g1u, g2u, embed_item, g1i, g2i, user, item_i, item_j, acc);

    lrgccf_finalize<<<1, 1, 0, stream>>>(acc, out);
}